// GraphConv2d_39762807226774
// MI455X (gfx1250) — compile-verified
//
#include <hip/hip_runtime.h>

typedef __attribute__((ext_vector_type(2))) float v2f;
typedef __attribute__((ext_vector_type(4))) float v4f;
typedef __attribute__((ext_vector_type(8))) float v8f;

#define BB   8
#define CC   64
#define NN   4096
#define OO   64
#define KNN  20

// ---------------------------------------------------------------------------
// Kernel 1: transpose x (B,C,N) -> pts (B,N,C) and compute sq norms per point
// ---------------------------------------------------------------------------
__global__ void __launch_bounds__(256)
transpose_sq_kernel(const float* __restrict__ x,
                    float* __restrict__ pts,
                    float* __restrict__ sq) {
  __shared__ float tile[64][65];
  const int b  = blockIdx.y;
  const int n0 = blockIdx.x * 64;
  const int t  = threadIdx.x;
  const int nn = t & 63;
  const int cg = t >> 6;            // 0..3
#pragma unroll
  for (int i = 0; i < 16; ++i) {
    int c = cg * 16 + i;
    tile[c][nn] = x[((size_t)b * CC + c) * NN + n0 + nn];   // coalesced along n
  }
  __syncthreads();
  const int cc = t & 63;
#pragma unroll
  for (int i = 0; i < 16; ++i) {
    int n = cg * 16 + i;
    pts[((size_t)b * NN + n0 + n) * CC + cc] = tile[cc][n]; // coalesced along c
  }
  if (t < 64) {
    float s = 0.f;
#pragma unroll
    for (int c = 0; c < 64; ++c) { float v = tile[c][t]; s += v * v; }
    sq[(size_t)b * NN + n0 + t] = s;
  }
}

// ---------------------------------------------------------------------------
// Kernel 2: Y1 = pts * W1^T ; Yc = pts * (W2-W1)^T + bias   (WMMA f32 16x16x4)
// One wave -> 16 rows x 64 cols of both outputs. W staged in padded LDS.
// ---------------------------------------------------------------------------
__global__ void __launch_bounds__(256)
feat_wmma_kernel(const float* __restrict__ pts,
                 const float* __restrict__ W,
                 const float* __restrict__ bias,
                 float* __restrict__ Y1,
                 float* __restrict__ Yc) {
  __shared__ __align__(16) float w1[64][68];
  __shared__ __align__(16) float wd[64][68];
  const int t = threadIdx.x;
  for (int i = t; i < 64 * 64; i += 256) {
    int o = i >> 6, c = i & 63;
    float a = W[o * 128 + c];
    w1[o][c] = a;
    wd[o][c] = W[o * 128 + 64 + c] - a;
  }
  __syncthreads();

  const int lane = t & 31;
  const int wave = t >> 5;
  const int r    = lane & 15;       // WMMA M/N index within tile
  const int hi   = lane >> 4;       // selects K pair (0 -> K0,K1 ; 1 -> K2,K3)
  const size_t row0 = (size_t)blockIdx.x * 128 + (size_t)wave * 16;
  const float* arow = pts + (row0 + r) * CC;

  v8f acc1[4] = {};
  v8f accc[4] = {};
#pragma unroll
  for (int kk = 0; kk < 16; ++kk) {
    const int k0 = kk * 4 + hi * 2;
    v2f a;
    a.x = arow[k0];
    a.y = arow[k0 + 1];
#pragma unroll
    for (int ot = 0; ot < 4; ++ot) {
      const int o = ot * 16 + r;
      v2f b1 = *(const v2f*)&w1[o][k0];   // conflict-free: bank = (4r+k0)%64
      v2f bd = *(const v2f*)&wd[o][k0];
      acc1[ot] = __builtin_amdgcn_wmma_f32_16x16x4_f32(
          false, a, false, b1, (short)0, acc1[ot], false, false);
      accc[ot] = __builtin_amdgcn_wmma_f32_16x16x4_f32(
          false, a, false, bd, (short)0, accc[ot], false, false);
    }
  }
#pragma unroll
  for (int ot = 0; ot < 4; ++ot) {
    const int o = ot * 16 + r;
    const float bv = bias[o];
#pragma unroll
    for (int v = 0; v < 8; ++v) {
      const size_t m = row0 + v + 8 * hi;   // C/D layout: M = v + 8*(lane>=16)
      Y1[m * OO + o] = acc1[ot][v];
      Yc[m * OO + o] = accc[ot][v] + bv;
    }
  }
}

// ---------------------------------------------------------------------------
// Kernel 3: fused KNN — WMMA distance tiles + per-lane register top-20.
// Double-buffered candidate staging via GLOBAL_LOAD_ASYNC_TO_LDS_B128
// (ASYNCcnt path, no VGPR round-trip); 2 candidate tiles x 2 K-split
// accumulators = 4 independent WMMA chains per round.
// ---------------------------------------------------------------------------
__device__ __forceinline__ void async_stage_tile(const float* __restrict__ src_base,
                                                 float (*dst)[68],
                                                 int t) {
  // 32 rows x 64 floats = 8 KB; 256 threads x 2 x b128
  const int rr = t >> 3;          // 0..31
  const int cc = (t & 7) * 8;     // 0..56
  const float* src = src_base + (size_t)rr * CC + cc;
  unsigned lds0 = (unsigned)(size_t)&dst[rr][cc];      // low 32 bits of flat LDS
  asm volatile("global_load_async_to_lds_b128 %0, %1, off"
               :: "v"(lds0), "v"(src) : "memory");
  asm volatile("global_load_async_to_lds_b128 %0, %1, off offset:16"
               :: "v"(lds0), "v"(src) : "memory");
}

__global__ void __launch_bounds__(256)
knn_wmma_kernel(const float* __restrict__ pts,
                const float* __restrict__ sq,
                int* __restrict__ knn_idx) {
  __shared__ __align__(16) float ctile[2][32][68];  // padded: conflict-free frags
  __shared__ float mval[8][32][KNN];
  __shared__ int   midx[8][32][KNN];

  const int b    = blockIdx.y;
  const int t    = threadIdx.x;
  const int lane = t & 31;
  const int wave = t >> 5;
  const int r    = lane & 15;
  const int hi   = lane >> 4;
  const int q    = blockIdx.x * 128 + wave * 16 + r;   // this lane's query column
  const float* base = pts + (size_t)b * NN * CC;
  const float* sqb  = sq + (size_t)b * NN;

  // Preload query B-fragments for the whole K=64 (16 k-steps) into registers.
  v2f bf[16];
  const float* qrow = base + (size_t)q * CC;
#pragma unroll
  for (int kk = 0; kk < 16; ++kk) {
    const int k0 = kk * 4 + hi * 2;
    bf[kk].x = qrow[k0];
    bf[kk].y = qrow[k0 + 1];
  }
  const float sq_q = sqb[q];

  float tv[KNN];
  int   ti[KNN];
#pragma unroll
  for (int j = 0; j < KNN; ++j) { tv[j] = -3.402823466e38f; ti[j] = 0; }

  // Prologue: kick off tile 0.
  async_stage_tile(base, ctile[0], t);

  for (int mt = 0; mt < NN / 32; ++mt) {
    const int m0  = mt * 32;
    const int buf = mt & 1;

    // Candidate sq-norms: 8+8 contiguous floats per lane, straight from L2.
    v4f sqa0 = *(const v4f*)(sqb + m0 + 8 * hi);
    v4f sqa1 = *(const v4f*)(sqb + m0 + 8 * hi + 4);
    v4f sqb0 = *(const v4f*)(sqb + m0 + 16 + 8 * hi);
    v4f sqb1 = *(const v4f*)(sqb + m0 + 16 + 8 * hi + 4);

    asm volatile("s_wait_asynccnt 0" ::: "memory");   // tile mt resident
    __syncthreads();
    if (mt + 1 < NN / 32)                              // overlap: stage mt+1
      async_stage_tile(base + (size_t)(m0 + 32) * CC, ctile[buf ^ 1], t);

    v8f acc0a = {}, acc0b = {}, acc1a = {}, acc1b = {};
#pragma unroll
    for (int kk = 0; kk < 16; kk += 2) {
      const int k0 = kk * 4 + hi * 2;
      const int k1 = k0 + 4;
      v2f a0 = *(const v2f*)&ctile[buf][r][k0];
      v2f a1 = *(const v2f*)&ctile[buf][16 + r][k0];
      v2f a2 = *(const v2f*)&ctile[buf][r][k1];
      v2f a3 = *(const v2f*)&ctile[buf][16 + r][k1];
      acc0a = __builtin_amdgcn_wmma_f32_16x16x4_f32(
          false, a0, false, bf[kk], (short)0, acc0a, false, false);
      acc1a = __builtin_amdgcn_wmma_f32_16x16x4_f32(
          false, a1, false, bf[kk], (short)0, acc1a, false, false);
      acc0b = __builtin_amdgcn_wmma_f32_16x16x4_f32(
          false, a2, false, bf[kk + 1], (short)0, acc0b, false, false);
      acc1b = __builtin_amdgcn_wmma_f32_16x16x4_f32(
          false, a3, false, bf[kk + 1], (short)0, acc1b, false, false);
    }

    // top-20 update: 16 candidates per lane this round
#pragma unroll
    for (int v = 0; v < 8; ++v) {
      const int   m     = m0 + v + 8 * hi;
      const float sc    = (v < 4) ? sqa0[v & 3] : sqa1[v & 3];
      const float negd2 = 2.f * (acc0a[v] + acc0b[v]) - sq_q - sc;
      if (negd2 > tv[0]) {
        tv[0] = negd2; ti[0] = m;
#pragma unroll
        for (int j = 0; j < KNN - 1; ++j) {
          if (tv[j] > tv[j + 1]) {
            float tf = tv[j]; tv[j] = tv[j + 1]; tv[j + 1] = tf;
            int   tn = ti[j]; ti[j] = ti[j + 1]; ti[j + 1] = tn;
          }
        }
      }
    }
#pragma unroll
    for (int v = 0; v < 8; ++v) {
      const int   m     = m0 + 16 + v + 8 * hi;
      const float sc    = (v < 4) ? sqb0[v & 3] : sqb1[v & 3];
      const float negd2 = 2.f * (acc1a[v] + acc1b[v]) - sq_q - sc;
      if (negd2 > tv[0]) {
        tv[0] = negd2; ti[0] = m;
#pragma unroll
        for (int j = 0; j < KNN - 1; ++j) {
          if (tv[j] > tv[j + 1]) {
            float tf = tv[j]; tv[j] = tv[j + 1]; tv[j + 1] = tf;
            int   tn = ti[j]; ti[j] = ti[j + 1]; ti[j + 1] = tn;
          }
        }
      }
    }
  }

  // Lane pair (L, L+16) each hold top-20 of disjoint candidate subsets: merge.
#pragma unroll
  for (int j = 0; j < KNN; ++j) {
    mval[wave][lane][j] = tv[j];
    midx[wave][lane][j] = ti[j];
  }
  __syncthreads();
  if (hi == 0) {
    int ia = KNN - 1, ib = KNN - 1;
    int* outp = knn_idx + ((size_t)b * NN + q) * KNN;
#pragma unroll
    for (int s = 0; s < KNN; ++s) {
      const float av = mval[wave][lane][ia];
      const int   ai = midx[wave][lane][ia];
      const float bv = mval[wave][lane + 16][ib];
      const int   bi = midx[wave][lane + 16][ib];
      const bool takeA = (av > bv) || (av == bv && ai < bi);  // lax.top_k tie-break
      outp[s] = takeA ? ai : bi;
      if (takeA) --ia; else --ib;
    }
  }
}

// ---------------------------------------------------------------------------
// Kernel 4: out[b,o,n] = max_k Y1[b, idx[b,n,k], o] + Yc[b,n,o]
// 64 threads per point -> each gather is a coalesced 256B read.
// ---------------------------------------------------------------------------
__global__ void __launch_bounds__(256)
gather_max_kernel(const int* __restrict__ knn_idx,
                  const float* __restrict__ Y1,
                  const float* __restrict__ Yc,
                  float* __restrict__ out) {
  const int t = threadIdx.x;
  const int p = blockIdx.x * 4 + (t >> 6);    // point id in [0, B*N)
  const int o = t & 63;
  const int b = p >> 12;
  const int n = p & (NN - 1);
  const int* ip = knn_idx + (size_t)p * KNN;
  const float* y = Y1 + (size_t)b * NN * OO;
  float mv = -3.402823466e38f;
#pragma unroll
  for (int k = 0; k < KNN; ++k) {
    const int m = ip[k];
    mv = fmaxf(mv, y[(size_t)m * OO + o]);
  }
  out[((size_t)b * OO + o) * NN + n] = mv + Yc[(size_t)p * OO + o];
}

// ---------------------------------------------------------------------------
extern "C" void kernel_launch(void* const* d_in, const int* in_sizes, int n_in,
                              void* d_out, int out_size, void* d_ws, size_t ws_size,
                              hipStream_t stream) {
  const float* x    = (const float*)d_in[0];   // (8, 64, 4096)
  const float* W    = (const float*)d_in[1];   // (64, 128)
  const float* bias = (const float*)d_in[2];   // (64,)
  float* out = (float*)d_out;                  // (8, 64, 4096)

  char* ws = (char*)d_ws;
  size_t off = 0;
  float* pts = (float*)(ws + off); off += (size_t)BB * NN * CC * sizeof(float);  // 8 MB
  float* sq  = (float*)(ws + off); off += (size_t)BB * NN * sizeof(float);       // 128 KB
  float* Y1  = (float*)(ws + off); off += (size_t)BB * NN * OO * sizeof(float);  // 8 MB
  float* Yc  = (float*)(ws + off); off += (size_t)BB * NN * OO * sizeof(float);  // 8 MB
  int*   idx = (int*)  (ws + off);                                               // 2.5 MB

  transpose_sq_kernel<<<dim3(NN / 64, BB), 256, 0, stream>>>(x, pts, sq);
  feat_wmma_kernel<<<dim3((BB * NN) / 128), 256, 0, stream>>>(pts, W, bias, Y1, Yc);
  knn_wmma_kernel<<<dim3(NN / 128, BB), 256, 0, stream>>>(pts, sq, idx);
  gather_max_kernel<<<dim3((BB * NN) / 4), 256, 0, stream>>>(idx, Y1, Yc, out);
}